// PSP_Pooling_52639119180499
// MI455X (gfx1250) — compile-verified
//
#include <hip/hip_runtime.h>

typedef __attribute__((ext_vector_type(2))) float v2f;
typedef __attribute__((ext_vector_type(8))) float v8f;

#define NB 4
#define NC 256
#define NH 192
#define NW 192
#define HW (NH * NW)          // 36864
#define EPS 1e-5f

// scratch layout (float offsets)
#define OFF_M     0           // [b][c][85] pyramid maxima
#define OFF_Y     87040       // [b][o][85] conv outputs -> normalized p
#define OFF_S     174080      // [b][o2][64] block-constant additive term
#define OFF_DSUM  239616      // [4][256]
#define OFF_DSQ   240640      // [4][256]
#define OFF_FSUM  241664      // [256]
#define OFF_FSQ   241920      // [256]
#define STATS_CNT 2560

__device__ __forceinline__ int lvl_off(int d) {
    return (d == 0) ? 0 : (d == 1) ? 1 : (d == 2) ? 5 : 21;
}

// ---------------- K0: zero BN accumulators ----------------
__global__ void k0_zero(float* stats) {
    for (int i = threadIdx.x; i < STATS_CNT; i += 256) stats[i] = 0.0f;
}

// ---------------- K1: pyramid max pooling ----------------
__global__ void k1_pool(const float* __restrict__ x, float* __restrict__ mAll) {
    __shared__ float part[256];
    __shared__ float m8[64];
    __shared__ float m4[16];
    __shared__ float m2[4];
    const int bc  = blockIdx.x;
    const int tid = threadIdx.x;
    const int blk = tid >> 2, sub = tid & 3;
    const int gy = blk >> 3, gx = blk & 7;
    const float* xp = x + (size_t)bc * HW;

    float mx = -3.4e38f;
    const int r0 = gy * 24 + sub * 6;
    const int c0 = gx * 24;
    for (int r = 0; r < 6; ++r) {
        const float4* row = (const float4*)(xp + (size_t)(r0 + r) * NW + c0);
        #pragma unroll
        for (int q = 0; q < 6; ++q) {
            float4 v = row[q];
            mx = fmaxf(mx, fmaxf(fmaxf(v.x, v.y), fmaxf(v.z, v.w)));
        }
    }
    part[tid] = mx;
    __syncthreads();
    if (tid < 64) {
        m8[tid] = fmaxf(fmaxf(part[tid * 4], part[tid * 4 + 1]),
                        fmaxf(part[tid * 4 + 2], part[tid * 4 + 3]));
    }
    __syncthreads();
    if (tid < 16) {
        int y = tid >> 2, xx = tid & 3;
        m4[tid] = fmaxf(fmaxf(m8[(2 * y) * 8 + 2 * xx],     m8[(2 * y) * 8 + 2 * xx + 1]),
                        fmaxf(m8[(2 * y + 1) * 8 + 2 * xx], m8[(2 * y + 1) * 8 + 2 * xx + 1]));
    }
    __syncthreads();
    if (tid < 4) {
        int y = tid >> 1, xx = tid & 1;
        m2[tid] = fmaxf(fmaxf(m4[(2 * y) * 4 + 2 * xx],     m4[(2 * y) * 4 + 2 * xx + 1]),
                        fmaxf(m4[(2 * y + 1) * 4 + 2 * xx], m4[(2 * y + 1) * 4 + 2 * xx + 1]));
    }
    __syncthreads();
    float* out = mAll + (size_t)bc * 85;
    if (tid == 0) out[0] = fmaxf(fmaxf(m2[0], m2[1]), fmaxf(m2[2], m2[3]));
    if (tid < 4)  out[1 + tid] = m2[tid];
    if (tid < 16) out[5 + tid] = m4[tid];
    if (tid < 64) out[21 + tid] = m8[tid];
}

__device__ __forceinline__ void decode_job(int j, int& d, int& b, int& pos, int& n) {
    if (j < 4)        { d = 0; j -= 0;  n = 1; }
    else if (j < 20)  { d = 1; j -= 4;  n = 4; }
    else if (j < 84)  { d = 2; j -= 20; n = 16; }
    else              { d = 3; j -= 84; n = 64; }
    b = j / n; pos = j % n;
}

// ---------------- K2a: per-depth 1x1 conv at distinct columns ----------------
__global__ void k2a_conv(const float* __restrict__ mAll, const float* __restrict__ w,
                         float* __restrict__ y, float* __restrict__ dsum,
                         float* __restrict__ dsq) {
    __shared__ float mv[NC];
    int d, b, pos, n;
    decode_job(blockIdx.x, d, b, pos, n);
    const int off = lvl_off(d);
    const int o = threadIdx.x;
    mv[o] = mAll[(size_t)(b * NC + o) * 85 + off + pos];
    __syncthreads();
    const float* wr = w + (size_t)d * NC * NC + (size_t)o * NC;
    float acc = 0.0f;
    #pragma unroll 8
    for (int c = 0; c < NC; ++c) acc = fmaf(wr[c], mv[c], acc);
    y[(size_t)(b * NC + o) * 85 + off + pos] = acc;
    atomicAdd(&dsum[d * NC + o], acc);
    atomicAdd(&dsq[d * NC + o], acc * acc);
}

// ---------------- K2b: batchnorm the small conv outputs in place ----------------
__global__ void k2b_bn(float* __restrict__ y, const float* __restrict__ gs,
                       const float* __restrict__ bs, const float* __restrict__ dsum,
                       const float* __restrict__ dsq) {
    int d, b, pos, n;
    decode_job(blockIdx.x, d, b, pos, n);
    const int off = lvl_off(d);
    const int o = threadIdx.x;
    const float cnt = (float)(NB * n);
    const float mu = dsum[d * NC + o] / cnt;
    const float var = dsq[d * NC + o] / cnt - mu * mu;
    const size_t idx = (size_t)(b * NC + o) * 85 + off + pos;
    const float v = y[idx];
    y[idx] = (v - mu) * rsqrtf(var + EPS) * gs[d * NC + o] + bs[d * NC + o];
}

// ---------------- K2c: fold pyramid levels through wf -> S[b][o2][8x8] ----------------
__global__ void k2c_fold(const float* __restrict__ p, const float* __restrict__ wf,
                         float* __restrict__ S) {
    __shared__ float pv[4 * NC];
    const int b = blockIdx.x >> 6;
    const int pos8 = blockIdx.x & 63;
    const int gy8 = pos8 >> 3, gx8 = pos8 & 7;
    const int tid = threadIdx.x;
    #pragma unroll
    for (int d = 0; d < 4; ++d) {
        const int g = 1 << d;
        const int pd = (gy8 >> (3 - d)) * g + (gx8 >> (3 - d));
        pv[d * NC + tid] = p[(size_t)(b * NC + tid) * 85 + lvl_off(d) + pd];
    }
    __syncthreads();
    const int o2 = tid;
    float acc = 0.0f;
    #pragma unroll
    for (int d = 0; d < 4; ++d) {
        const float* wr = wf + (size_t)o2 * (5 * NC) + (1 + d) * NC;
        #pragma unroll 8
        for (int o = 0; o < NC; ++o) acc = fmaf(wr[o], pv[d * NC + o], acc);
    }
    S[(size_t)(b * NC + o2) * 64 + pos8] = acc;
}

// ---------------- K3: main GEMM via V_WMMA_F32_16X16X4_F32 ----------------
// block = 256 threads (8 waves); block tile: M=32 (2 A-frags) x N=512
// wave tile: 32(M) x 64(N); K = 256 in steps of 4; B frags reused for 2 M-tiles.
// blockIdx ordered with the 8 M-blocks innermost so blocks sharing an x tile
// launch adjacently and hit L2.
#define LDSW_STRIDE 258
__global__ void k3_gemm(const float* __restrict__ x, const float* __restrict__ wf,
                        const float* __restrict__ S, float* __restrict__ out,
                        float* __restrict__ fsum, float* __restrict__ fsq) {
    __shared__ float ldsW[32 * LDSW_STRIDE];
    __shared__ float lSum[32];
    __shared__ float lSq[32];

    const int gid  = blockIdx.x;
    const int mt   = gid & 7;          // fastest: M-block (shares x tile in L2)
    const int rest = gid >> 3;
    const int pblk = rest % 72;
    const int b    = rest / 72;
    const int m0 = mt * 32;
    const int pixBase = pblk * 512;
    const int tid = threadIdx.x;

    // stage wf row-block [m0..m0+32) x 256 into LDS (padded stride)
    for (int i = tid; i < 32 * NC; i += 256) {
        const int r = i >> 8, k = i & 255;
        ldsW[r * LDSW_STRIDE + k] = wf[(size_t)(m0 + r) * (5 * NC) + k];
    }
    if (tid < 32) { lSum[tid] = 0.0f; lSq[tid] = 0.0f; }
    __syncthreads();

    const int lane = tid & 31;
    const int wave = tid >> 5;
    const int half = lane >> 4;       // 0: lanes 0-15, 1: lanes 16-31
    const int lm   = lane & 15;
    const int pixWave = pixBase + wave * 64;
    const float* xb = x + (size_t)b * NC * HW;

    v8f acc[2][4] = {};
    for (int k0 = 0; k0 < NC; k0 += 4) {
        // A fragments: lanes 0-15 carry K=k0,k0+1; lanes 16-31 carry K=k0+2,k0+3
        v2f a0 = *(const v2f*)&ldsW[lm * LDSW_STRIDE + k0 + half * 2];
        v2f a1 = *(const v2f*)&ldsW[(16 + lm) * LDSW_STRIDE + k0 + half * 2];
        #pragma unroll
        for (int t = 0; t < 4; ++t) {
            const int n = pixWave + t * 16 + lm;
            v2f bf;
            bf.x = xb[(size_t)(k0 + half) * HW + n];       // K rows k0 / k0+1
            bf.y = xb[(size_t)(k0 + 2 + half) * HW + n];   // K rows k0+2 / k0+3
            acc[0][t] = __builtin_amdgcn_wmma_f32_16x16x4_f32(
                false, a0, false, bf, (short)0, acc[0][t], false, false);
            acc[1][t] = __builtin_amdgcn_wmma_f32_16x16x4_f32(
                false, a1, false, bf, (short)0, acc[1][t], false, false);
        }
    }

    // epilogue: add block-constant term, store pre-BN, accumulate channel stats
    float sumr[16], sqr[16];
    #pragma unroll
    for (int i = 0; i < 16; ++i) { sumr[i] = 0.0f; sqr[i] = 0.0f; }

    const float* Sb = S + ((size_t)(b * NC) << 6);
    #pragma unroll
    for (int t = 0; t < 4; ++t) {
        const int n = pixWave + t * 16 + lm;
        const int h = n / NW;
        const int w = n - h * NW;
        const int sblk = (h / 24) * 8 + (w / 24);   // invariant over v, mi
        #pragma unroll
        for (int mi = 0; mi < 2; ++mi) {
            #pragma unroll
            for (int v = 0; v < 8; ++v) {
                const int m = m0 + mi * 16 + v + half * 8;
                const float val = acc[mi][t][v] + Sb[((size_t)m << 6) + sblk];
                out[(size_t)(b * NC + m) * HW + n] = val;
                sumr[mi * 8 + v] += val;
                sqr[mi * 8 + v]  += val * val;
            }
        }
    }
    #pragma unroll
    for (int i = 0; i < 16; ++i) {
        const int mi = i >> 3, v = i & 7;
        const int slot = mi * 16 + v + half * 8;    // channel offset m - m0
        atomicAdd(&lSum[slot], sumr[i]);
        atomicAdd(&lSq[slot],  sqr[i]);
    }
    __syncthreads();
    if (tid < 32) {
        atomicAdd(&fsum[m0 + tid], lSum[tid]);
        atomicAdd(&fsq[m0 + tid],  lSq[tid]);
    }
}

// ---------------- K4: final batchnorm, in place on d_out ----------------
__global__ void k4_bn(float* __restrict__ out, const float* __restrict__ fsum,
                      const float* __restrict__ fsq, const float* __restrict__ gf,
                      const float* __restrict__ bf) {
    const size_t idx = (size_t)blockIdx.x * 256 + threadIdx.x;
    const int c = (int)((idx / HW) & (NC - 1));
    const float cnt = (float)((size_t)NB * HW);
    const float mu = fsum[c] / cnt;
    const float var = fsq[c] / cnt - mu * mu;
    const float v = out[idx];
    out[idx] = (v - mu) * rsqrtf(var + EPS) * gf[c] + bf[c];
}

extern "C" void kernel_launch(void* const* d_in, const int* in_sizes, int n_in,
                              void* d_out, int out_size, void* d_ws, size_t ws_size,
                              hipStream_t stream) {
    const float* x  = (const float*)d_in[0];
    const float* ws = (const float*)d_in[1];
    const float* gs = (const float*)d_in[2];
    const float* bs = (const float*)d_in[3];
    const float* wf = (const float*)d_in[4];
    const float* gf = (const float*)d_in[5];
    const float* bf = (const float*)d_in[6];
    float* out = (float*)d_out;
    float* w = (float*)d_ws;

    float* mAll  = w + OFF_M;
    float* yv    = w + OFF_Y;
    float* S     = w + OFF_S;
    float* dsum  = w + OFF_DSUM;
    float* dsq   = w + OFF_DSQ;
    float* fsum  = w + OFF_FSUM;
    float* fsq   = w + OFF_FSQ;

    k0_zero<<<1, 256, 0, stream>>>(dsum);  // zeroes all STATS_CNT floats from dsum
    k1_pool<<<NB * NC, 256, 0, stream>>>(x, mAll);
    k2a_conv<<<340, 256, 0, stream>>>(mAll, ws, yv, dsum, dsq);
    k2b_bn<<<340, 256, 0, stream>>>(yv, gs, bs, dsum, dsq);
    k2c_fold<<<NB * 64, 256, 0, stream>>>(yv, wf, S);
    k3_gemm<<<NB * 72 * 8, 256, 0, stream>>>(x, wf, S, out, fsum, fsq);
    k4_bn<<<(NB * NC * HW) / 256, 256, 0, stream>>>(out, fsum, fsq, gf, bf);
}